// GraphSAGEModel_26216480375160
// MI455X (gfx1250) — compile-verified
//
#include <hip/hip_runtime.h>

// ---------------------------------------------------------------------------
// GraphSAGE 2-layer (mean aggr) for MI455X / gfx1250.
//   N=100000 nodes, E=1600000 edges, F_IN=32, HID=16.
// Strategy:
//   * scatter-add aggregation with global_atomic_add_f32 (accumulators live in
//     the 192MB L2 -> atomics execute at L2 rate)
//   * per-node linear layers with full-precision V_WMMA_F32_16X16X4_F32
// d_out doubles as the layer-1 sum accumulator (exactly N*32 floats).
// ---------------------------------------------------------------------------

#define NNODES 100000
#define NEDGES 1600000
#define F_IN   32
#define HID    16

typedef float v2f __attribute__((ext_vector_type(2)));
typedef float v8f __attribute__((ext_vector_type(8)));

// ---------------- zero fill ----------------
__global__ void k_zero(float* __restrict__ p, int n) {
    int i = blockIdx.x * blockDim.x + threadIdx.x;
    if (i < n) p[i] = 0.0f;
}

// ---------------- layer-1 scatter: sum x[src] into acc[dst], count deg ------
// 8 threads per edge, 4 features each (float4 gather, 4 f32 atomics).
__global__ void k_scatter32(const float* __restrict__ x,
                            const long long* __restrict__ ei,
                            float* __restrict__ acc,
                            float* __restrict__ deg) {
    unsigned t = blockIdx.x * blockDim.x + threadIdx.x;
    unsigned e = t >> 3;
    unsigned c = t & 7u;
    if (e >= NEDGES) return;
    long long src = ei[e];
    long long dst = ei[NEDGES + e];
    const float4 v = *(const float4*)(x + src * F_IN + c * 4);
    float* a = acc + dst * F_IN + c * 4;
    atomicAdd(a + 0, v.x);
    atomicAdd(a + 1, v.y);
    atomicAdd(a + 2, v.z);
    atomicAdd(a + 3, v.w);
    if (c == 0) atomicAdd(deg + dst, 1.0f);
}

// ---------------- layer-2 scatter: sum h[src] into acc2[dst] ----------------
__global__ void k_scatter16(const float* __restrict__ h,
                            const long long* __restrict__ ei,
                            float* __restrict__ acc) {
    unsigned t = blockIdx.x * blockDim.x + threadIdx.x;
    unsigned e = t >> 2;
    unsigned c = t & 3u;
    if (e >= NEDGES) return;
    long long src = ei[e];
    long long dst = ei[NEDGES + e];
    const float4 v = *(const float4*)(h + src * HID + c * 4);
    float* a = acc + dst * HID + c * 4;
    atomicAdd(a + 0, v.x);
    atomicAdd(a + 1, v.y);
    atomicAdd(a + 2, v.z);
    atomicAdd(a + 3, v.w);
}

// ---------------- dense layer 1: h = relu((sum/deg)@W_l^T + b + x@W_r^T) ----
// One wave (32 threads) per 16-node tile. Full-fp32 WMMA, K=32 as 8 chained
// 16x16x4 steps per term. A layout (ISA 7.12.2, 32-bit 16x4): lanes 0-15 hold
// M=lane,K={k0,k0+1}; lanes 16-31 hold M=lane-16,K={k0+2,k0+3}. B mirrored
// with lane%16 = N column. C/D: lane%16 = N, VGPR r -> M = r + 8*(lane/16).
__global__ void k_dense1(const float* __restrict__ x,
                         const float* __restrict__ aggsum,
                         const float* __restrict__ deg,
                         const float* __restrict__ W_l,   // [16,32]
                         const float* __restrict__ b_l,   // [16]
                         const float* __restrict__ W_r,   // [16,32]
                         float* __restrict__ h) {
    const int tile  = blockIdx.x;          // 6250 tiles of 16 nodes
    const int lane  = threadIdx.x;         // 0..31, full wave
    const int m0    = tile << 4;
    const int mrow  = m0 + (lane & 15);
    const int khalf = (lane >> 4) << 1;    // 0 for lanes 0-15, 2 for 16-31
    const int n     = lane & 15;

    const float inv  = 1.0f / fmaxf(deg[mrow], 1.0f);
    const float bias = b_l[n];
    const float* arow = aggsum + mrow * F_IN;
    const float* xrow = x      + mrow * F_IN;

    v8f c;
#pragma unroll
    for (int r = 0; r < 8; ++r) c[r] = bias;

#pragma unroll
    for (int kk = 0; kk < 8; ++kk) {               // mean-aggregated term
        const int k0 = kk * 4 + khalf;
        v2f a, b;
        a.x = arow[k0] * inv;      a.y = arow[k0 + 1] * inv;
        b.x = W_l[n * F_IN + k0];  b.y = W_l[n * F_IN + k0 + 1];
        c = __builtin_amdgcn_wmma_f32_16x16x4_f32(false, a, false, b,
                                                  (short)0, c, false, false);
    }
#pragma unroll
    for (int kk = 0; kk < 8; ++kk) {               // root (self) term
        const int k0 = kk * 4 + khalf;
        v2f a, b;
        a.x = xrow[k0];            a.y = xrow[k0 + 1];
        b.x = W_r[n * F_IN + k0];  b.y = W_r[n * F_IN + k0 + 1];
        c = __builtin_amdgcn_wmma_f32_16x16x4_f32(false, a, false, b,
                                                  (short)0, c, false, false);
    }

    const int rbase = (lane >> 4) ? 8 : 0;
#pragma unroll
    for (int r = 0; r < 8; ++r)
        h[(m0 + rbase + r) * HID + n] = fmaxf(c[r], 0.0f);   // ReLU
}

// ---------------- dense layer 2: out = (sum2/deg)@W2_l^T + b2 + h@W2_r^T ----
// Grid = 6250 node-tiles x 2 output-column tiles; K=16 -> 4 WMMAs per term.
__global__ void k_dense2(const float* __restrict__ h,
                         const float* __restrict__ aggsum,   // [N,16]
                         const float* __restrict__ deg,
                         const float* __restrict__ W_l,      // [32,16]
                         const float* __restrict__ b_l,      // [32]
                         const float* __restrict__ W_r,      // [32,16]
                         float* __restrict__ out) {          // [N,32]
    const int tile  = blockIdx.x;          // node tile (6250)
    const int ntile = blockIdx.y;          // output column tile (0 or 1)
    const int lane  = threadIdx.x;
    const int m0    = tile << 4;
    const int mrow  = m0 + (lane & 15);
    const int khalf = (lane >> 4) << 1;
    const int n     = lane & 15;
    const int ncol  = (ntile << 4) + n;    // global output column 0..31

    const float inv  = 1.0f / fmaxf(deg[mrow], 1.0f);
    const float bias = b_l[ncol];
    const float* arow = aggsum + mrow * HID;
    const float* hrow = h      + mrow * HID;

    v8f c;
#pragma unroll
    for (int r = 0; r < 8; ++r) c[r] = bias;

#pragma unroll
    for (int kk = 0; kk < 4; ++kk) {               // mean-aggregated term
        const int k0 = kk * 4 + khalf;
        v2f a, b;
        a.x = arow[k0] * inv;       a.y = arow[k0 + 1] * inv;
        b.x = W_l[ncol * HID + k0]; b.y = W_l[ncol * HID + k0 + 1];
        c = __builtin_amdgcn_wmma_f32_16x16x4_f32(false, a, false, b,
                                                  (short)0, c, false, false);
    }
#pragma unroll
    for (int kk = 0; kk < 4; ++kk) {               // root (self) term
        const int k0 = kk * 4 + khalf;
        v2f a, b;
        a.x = hrow[k0];             a.y = hrow[k0 + 1];
        b.x = W_r[ncol * HID + k0]; b.y = W_r[ncol * HID + k0 + 1];
        c = __builtin_amdgcn_wmma_f32_16x16x4_f32(false, a, false, b,
                                                  (short)0, c, false, false);
    }

    const int rbase = (lane >> 4) ? 8 : 0;
#pragma unroll
    for (int r = 0; r < 8; ++r)
        out[(m0 + rbase + r) * F_IN + ncol] = c[r];
}

// ---------------------------------------------------------------------------
extern "C" void kernel_launch(void* const* d_in, const int* in_sizes, int n_in,
                              void* d_out, int out_size, void* d_ws, size_t ws_size,
                              hipStream_t stream) {
    const float*     x    = (const float*)d_in[0];
    const long long* ei   = (const long long*)d_in[1];   // int64 per reference
    const float*     W1_l = (const float*)d_in[2];
    const float*     b1_l = (const float*)d_in[3];
    const float*     W1_r = (const float*)d_in[4];
    const float*     W2_l = (const float*)d_in[5];
    const float*     b2_l = (const float*)d_in[6];
    const float*     W2_r = (const float*)d_in[7];
    float* out = (float*)d_out;

    // Workspace layout (bytes): deg[N] | h[N*16] | agg2[N*16]  = 13.2 MB
    char*  ws   = (char*)d_ws;
    float* deg  = (float*)(ws);
    float* h    = (float*)(ws + (size_t)NNODES * 4);
    float* agg2 = (float*)(ws + (size_t)NNODES * 4 + (size_t)NNODES * HID * 4);

    // d_out doubles as the layer-1 sum accumulator (N*32 floats).
    float* agg1 = out;

    // 1) zero accumulators (fresh every launch -> deterministic replay)
    k_zero<<<(NNODES * F_IN + 255) / 256, 256, 0, stream>>>(agg1, NNODES * F_IN);
    k_zero<<<(NNODES        + 255) / 256, 256, 0, stream>>>(deg,  NNODES);
    k_zero<<<(NNODES * HID  + 255) / 256, 256, 0, stream>>>(agg2, NNODES * HID);

    // 2) layer-1 aggregation: 8 threads/edge
    k_scatter32<<<(NEDGES * 8) / 256, 256, 0, stream>>>(x, ei, agg1, deg);

    // 3) layer-1 dense (WMMA f32) + ReLU -> h
    k_dense1<<<NNODES / 16, 32, 0, stream>>>(x, agg1, deg, W1_l, b1_l, W1_r, h);

    // 4) layer-2 aggregation: 4 threads/edge
    k_scatter16<<<(NEDGES * 4) / 256, 256, 0, stream>>>(h, ei, agg2);

    // 5) layer-2 dense (WMMA f32) -> out
    k_dense2<<<dim3(NNODES / 16, 2), 32, 0, stream>>>(h, agg2, deg, W2_l, b2_l, W2_r, out);
}